// PullbackMetric_82446192214176
// MI455X (gfx1250) — compile-verified
//
#include <hip/hip_runtime.h>

typedef float v2f __attribute__((ext_vector_type(2)));
typedef float v8f __attribute__((ext_vector_type(8)));

#define GRID 512
#define GMASK (GRID - 1)

// per-wave LDS slab (floats); every region offset is 16B-aligned
#define L_BD   0     // 256 : blockdiag(Sinv0, Sinv1) 16x16 row-major, row r at r*16
#define L_DSIG 256   // 256 : dsig, cell h at h*128, axis m at m*64 (symmetric 8x8)
#define L_DMU  512   // 32  : dmu,  cell h at h*16,  axis m at m*8
#define L_PT   544   // 256 : P^T,  cell h at h*128, PT[c][v] = P[v][c], row c at c*16
#define SLAB   800   // 3200 B per wave

// wave-scope scheduling barrier: all LDS sharing here is intra-wave (each wave
// owns its slab) and same-wave LDS ops execute in order, so no HW barrier needed.
#if __has_builtin(__builtin_amdgcn_wave_barrier)
#define WAVE_SYNC() __builtin_amdgcn_wave_barrier()
#else
#define WAVE_SYNC() __syncthreads()
#endif

// butterfly add via ds_swizzle (group-of-32: and=0x1f, or=0, xor=off) — one DS
// op per step, no lane-index VALU.
#define SWZ_XOR_F(v, off) \
    __int_as_float(__builtin_amdgcn_ds_swizzle(__float_as_int(v), (((off) << 10) | 0x1f)))

__global__ __launch_bounds__(256)
void pullback_metric_kernel(const float* __restrict__ mu,
                            const float* __restrict__ sigma,
                            const float* __restrict__ sigma_inv,
                            float* __restrict__ out) {
    __shared__ float lds[8 * SLAB];
    const int lane = threadIdx.x & 31;
    const int wave = threadIdx.x >> 5;
    float* S = &lds[wave * SLAB];

    const int h   = lane >> 4;     // half-wave -> which of the 2 cells
    const int idx = lane & 15;

    const int pair = blockIdx.x * 8 + wave;   // this wave's cell pair
    const int cell = pair * 2 + h;            // this half-wave's cell
    const int x = cell >> 9, y = cell & GMASK;
    const int xp = (x + 1) & GMASK, xm = (x - 1) & GMASK;
    const int yp = (y + 1) & GMASK, ym = (y - 1) & GMASK;

    const size_t bC  = (size_t)(x  * GRID + y ) * 64;
    const size_t bXp = (size_t)(xp * GRID + y ) * 64;
    const size_t bXm = (size_t)(xm * GRID + y ) * 64;
    const size_t bYp = (size_t)(x  * GRID + yp) * 64;
    const size_t bYm = (size_t)(x  * GRID + ym) * 64;

    // ---- stage blockdiag(Sinv0,Sinv1) and dsig (b128 traffic, 16 lanes/cell)
    {
        // lane's float4 of sigma_inv = Sinv_h[row ri, cols c4..c4+3]
        const int ri = idx >> 1, c4 = (idx & 1) * 4;
        const int bdrow = L_BD + (h * 8 + ri) * 16;
        float4 v = ((const float4*)(sigma_inv + bC))[idx];
        *(float4*)&S[bdrow + h * 8 + c4]       = v;                        // diag
        *(float4*)&S[bdrow + (1 - h) * 8 + c4] = make_float4(0.f, 0.f, 0.f, 0.f); // off-diag

        float4 p = ((const float4*)(sigma + bXp))[idx];
        float4 q = ((const float4*)(sigma + bXm))[idx];
        float4 d;
        d.x = (p.x - q.x) * 0.5f; d.y = (p.y - q.y) * 0.5f;
        d.z = (p.z - q.z) * 0.5f; d.w = (p.w - q.w) * 0.5f;
        *(float4*)&S[L_DSIG + h * 128 + 4 * idx] = d;

        p = ((const float4*)(sigma + bYp))[idx];
        q = ((const float4*)(sigma + bYm))[idx];
        d.x = (p.x - q.x) * 0.5f; d.y = (p.y - q.y) * 0.5f;
        d.z = (p.z - q.z) * 0.5f; d.w = (p.w - q.w) * 0.5f;
        *(float4*)&S[L_DSIG + h * 128 + 64 + 4 * idx] = d;
    }
    // ---- dmu: every lane produces one central difference (2 cells x 2 axes x 8)
    {
        const int dd = idx & 7, ax = idx >> 3;
        const size_t ip = ax ? ((size_t)(x  * GRID + yp) * 8 + dd)
                             : ((size_t)(xp * GRID + y ) * 8 + dd);
        const size_t im = ax ? ((size_t)(x  * GRID + ym) * 8 + dd)
                             : ((size_t)(xm * GRID + y ) * 8 + dd);
        S[L_DMU + h * 16 + idx] = (mu[ip] - mu[im]) * 0.5f;
    }
    WAVE_SYNC();

    // ---- A operand: rows of blockdiag(Sinv0,Sinv1) read straight from LDS.
    // A-layout: lane r=lane&15 holds row r; lane half hh selects K sub-pair.
    const int r  = idx;
    const int hh = h;
    const int ab = L_BD + r * 16 + 2 * hh;
    const v2f a0 = *(const v2f*)&S[ab];        // K {0,1}+2hh
    const v2f a1 = *(const v2f*)&S[ab + 4];    // K {4,5}+2hh
    const v2f a2 = *(const v2f*)&S[ab + 8];    // K {8,9}+2hh
    const v2f a3 = *(const v2f*)&S[ab + 12];   // K {12,13}+2hh

    // ---- B operand: X rows 0-7 = [dsig0|dsig1] of cell0, rows 8-15 = cell1.
    // column c = r: axis m = c>>3, entry j = c&7; X[k,c] = dsig_m[k,j] = dsig_m[j,k]
    // (dsig symmetric) -> contiguous b64 reads of row j.
    const int xb0 = L_DSIG + (r >> 3) * 64 + (r & 7) * 8 + 2 * hh;
    const int xb1 = xb0 + 128;
    const v2f b0 = *(const v2f*)&S[xb0];       // X[2hh..2hh+1 , c]
    const v2f b1 = *(const v2f*)&S[xb0 + 4];   // X[4+2hh..    , c]
    const v2f b2 = *(const v2f*)&S[xb1];       // X[8+2hh..    , c]
    const v2f b3 = *(const v2f*)&S[xb1 + 4];   // X[12+2hh..   , c]

    // ---- D(16x16): rows 0-7 = Sinv0 @ X0, rows 8-15 = Sinv1 @ X1  (K=16)
    v8f acc = {};
    acc = __builtin_amdgcn_wmma_f32_16x16x4_f32(false, a0, false, b0, (short)0, acc, false, false);
    acc = __builtin_amdgcn_wmma_f32_16x16x4_f32(false, a1, false, b1, (short)0, acc, false, false);
    acc = __builtin_amdgcn_wmma_f32_16x16x4_f32(false, a2, false, b2, (short)0, acc, false, false);
    acc = __builtin_amdgcn_wmma_f32_16x16x4_f32(false, a3, false, b3, (short)0, acc, false, false);

    // ---- spill P transposed: each lane holds a COLUMN of P (acc[v] = P[v][c]),
    // so PT[c][0..7] is contiguous -> two b128 stores, no divergence.
    {
        const int pt = L_PT + h * 128 + idx * 16;
        *(float4*)&S[pt]     = make_float4(acc[0], acc[1], acc[2], acc[3]);
        *(float4*)&S[pt + 4] = make_float4(acc[4], acc[5], acc[6], acc[7]);
    }
    WAVE_SYNC();

    // ---- per-half-wave reduction over the 64 (i,k) pairs of its own cell:
    //   cov[m,n] = sum P[i,8m+k] * P[k,8n+i]
    //   mean[m,n] = sum dmu_m[i] * Sinv[i,k] * dmu_n[k]
    // lane covers i = idx>>1, k in [k0, k0+4), k0 = (idx&1)*4  (contiguous).
    {
        const int Pb = L_PT + h * 128;
        const int Mb = L_DMU + h * 16;
        const int i  = idx >> 1;
        const int k0 = (idx & 1) * 4;

        // P[k,i] = PT[i][k], P[k,8+i] = PT[8+i][k] : contiguous b128 rows of PT
        const float4 Pki0 = *(const float4*)&S[Pb + i * 16 + k0];
        const float4 Pki1 = *(const float4*)&S[Pb + (8 + i) * 16 + k0];
        // P[i,k] = PT[k][i], P[i,8+k] = PT[8+k][i] : strided (2addr pairs)
        float Pik0[4], Pik1[4];
#pragma unroll
        for (int t = 0; t < 4; ++t) {
            Pik0[t] = S[Pb + (k0 + t) * 16 + i];
            Pik1[t] = S[Pb + (8 + k0 + t) * 16 + i];
        }
        // Sinv_h[i, k0..] from the diagonal block of BD (row stride 16)
        const float4 sv  = *(const float4*)&S[L_BD + (h * 8 + i) * 16 + h * 8 + k0];
        const float  mi0 = S[Mb + i], mi1 = S[Mb + 8 + i];
        const float4 mk0 = *(const float4*)&S[Mb + k0];
        const float4 mk1 = *(const float4*)&S[Mb + 8 + k0];

        float c00 = 0.f, c01 = 0.f, c11 = 0.f;
        float e00 = 0.f, e01 = 0.f, e11 = 0.f;
#pragma unroll
        for (int t = 0; t < 4; ++t) {
            const float pik0 = Pik0[t];
            const float pik1 = Pik1[t];
            c00 += pik0 * ((const float*)&Pki0)[t];
            c01 += pik0 * ((const float*)&Pki1)[t];
            c11 += pik1 * ((const float*)&Pki1)[t];
            const float s  = ((const float*)&sv)[t];
            const float t0 = s * mi0, t1 = s * mi1;
            e00 += t0 * ((const float*)&mk0)[t];
            e01 += t0 * ((const float*)&mk1)[t];
            e11 += t1 * ((const float*)&mk1)[t];
        }
        float g00 = e00 + 0.5f * c00;
        float g01 = e01 + 0.5f * c01;   // == G[1,0] (mirrored upper triangle)
        float g11 = e11 + 0.5f * c11;

        // width-16 butterfly (xor 8,4,2,1 stays inside each 16-lane half)
        g00 += SWZ_XOR_F(g00, 8); g01 += SWZ_XOR_F(g01, 8); g11 += SWZ_XOR_F(g11, 8);
        g00 += SWZ_XOR_F(g00, 4); g01 += SWZ_XOR_F(g01, 4); g11 += SWZ_XOR_F(g11, 4);
        g00 += SWZ_XOR_F(g00, 2); g01 += SWZ_XOR_F(g01, 2); g11 += SWZ_XOR_F(g11, 2);
        g00 += SWZ_XOR_F(g00, 1); g01 += SWZ_XOR_F(g01, 1); g11 += SWZ_XOR_F(g11, 1);

        if (idx == 0) {   // lane 0 -> cell0, lane 16 -> cell1
            float4* o = (float4*)(out + (size_t)cell * 4);
            *o = make_float4(g00, g01, g01, g11);
        }
    }
}

extern "C" void kernel_launch(void* const* d_in, const int* in_sizes, int n_in,
                              void* d_out, int out_size, void* d_ws, size_t ws_size,
                              hipStream_t stream) {
    const float* mu        = (const float*)d_in[0];
    const float* sigma     = (const float*)d_in[1];
    const float* sigma_inv = (const float*)d_in[2];
    float* out             = (float*)d_out;

    const int blocks = (GRID * GRID) / 16;  // 8 waves/block, 2 cells/wave
    pullback_metric_kernel<<<blocks, 256, 0, stream>>>(mu, sigma, sigma_inv, out);
}